// PSDLoss_59536836657442
// MI455X (gfx1250) — compile-verified
//
#include <hip/hip_runtime.h>
#include <math.h>
#include <stdint.h>

// ---------------------------------------------------------------------------
// PSD radial profile on MI455X (gfx1250).
// rfft2 realized as DFT-by-GEMM so the f32 WMMA units do the heavy lifting:
//   F1 = G @ W      (G: [512x512] real luma, W: [512x288] complex rfft matrix,
//                    cols 257..287 zero padding so N tiles by 32)
//   F  = D @ F1     (D: [512x512] complex full DFT matrix)
// then power spectrum + radial segment sum + per-batch normalization.
// Memory floor: 201 MB pred reads ~ 9us @ 23.3 TB/s; GEMM work ~55 GFLOP f32,
// comfortably inside f32-WMMA throughput -> keep full f32 precision.
// Round 2/3: global->LDS tile fills use gfx1250 GLOBAL_LOAD_ASYNC_TO_LDS_B128
// (ASYNCcnt-tracked, no VGPR staging) with ping-pong LDS double buffering so
// the next K-tile's fetch overlaps the current tile's WMMA chain.
// ---------------------------------------------------------------------------

#define HH     512
#define WW     512
#define WC     257          // one-sided columns
#define NP     288          // padded N (multiple of 32)
#define NB     64           // batch
#define CHUNK  8            // batches in flight (ws ~31 MB)
#define VLEN   363          // radial bins
#define TWO_PI 6.283185307179586f
#define APAD   36           // LDS row stride: 16B-aligned, conflict-free cols

// workspace layout (float offsets)
#define OFF_WR    0u                          // 512*288
#define OFF_WI    (OFF_WR   + 512u*NP)
#define OFF_DR    (OFF_WI   + 512u*NP)        // 512*512
#define OFF_DI    (OFF_DR   + 512u*512u)
#define OFF_GRAY  (OFF_DI   + 512u*512u)      // CHUNK*512*512
#define OFF_F1R   (OFF_GRAY + (unsigned)CHUNK*512u*512u)
#define OFF_F1I   (OFF_F1R  + (unsigned)CHUNK*512u*NP)
#define OFF_FR    (OFF_F1I  + (unsigned)CHUNK*512u*NP)
#define OFF_FI    (OFF_FR   + (unsigned)CHUNK*512u*NP)
#define OFF_SUMS  (OFF_FI   + (unsigned)CHUNK*512u*NP)   // NB*VLEN
#define OFF_CNT   (OFF_SUMS + (unsigned)NB*VLEN)         // VLEN

typedef float v2f __attribute__((ext_vector_type(2)));
typedef float v4f __attribute__((ext_vector_type(4)));
typedef float v8f __attribute__((ext_vector_type(8)));
typedef int   v4i __attribute__((ext_vector_type(4)));

#if __has_builtin(__builtin_amdgcn_global_load_async_to_lds_b128)
#define ASYNC_LDS 1
typedef __attribute__((address_space(1))) v4i g4i;   // global int4
typedef __attribute__((address_space(3))) v4i l4i;   // LDS int4
#endif

// 16-byte global -> LDS copy. Async path: ASYNCcnt-tracked DMA, no VGPRs.
__device__ __forceinline__ void cp16(const float* g, float* l) {
#if defined(ASYNC_LDS)
    __builtin_amdgcn_global_load_async_to_lds_b128(
        (g4i*)(uintptr_t)g,
        (l4i*)(uint32_t)(uintptr_t)l,   // low 32 bits of generic LDS ptr = LDS offset
        0, 0);
#else
    *(v4f*)l = *(const v4f*)g;
#endif
}

__device__ __forceinline__ void wait_async() {
#if defined(ASYNC_LDS)
#if __has_builtin(__builtin_amdgcn_s_wait_asynccnt)
    __builtin_amdgcn_s_wait_asynccnt(0);
#else
    asm volatile("s_wait_asynccnt 0x0" ::: "memory");
#endif
#endif
}

// ---------------------------------------------------------------- twiddles --
__global__ void k_twiddle(float* __restrict__ ws) {
    int j = blockIdx.x * 16 + threadIdx.x;   // column
    int i = blockIdx.y * 16 + threadIdx.y;   // row
    if (i >= 512 || j >= 512) return;
    int m = (i * j) & 511;                   // exact angle reduction mod 512
    float th = -(TWO_PI / 512.0f) * (float)m;
    float sn, cs;
    __sincosf(th, &sn, &cs);
    ws[OFF_DR + (unsigned)i * 512u + j] = cs;
    ws[OFF_DI + (unsigned)i * 512u + j] = sn;
    if (j < NP) {
        bool live = (j < WC);
        ws[OFF_WR + (unsigned)i * NP + j] = live ? cs : 0.0f;
        ws[OFF_WI + (unsigned)i * NP + j] = live ? sn : 0.0f;
    }
}

// ------------------------------------------------------------------- init ---
__global__ void k_init(float* __restrict__ p, int n) {
    int i = blockIdx.x * 256 + threadIdx.x;
    if (i < n) p[i] = 0.0f;
}

__global__ void k_counts(const int* __restrict__ radius, float* __restrict__ cnt) {
    int i = blockIdx.x * 256 + threadIdx.x;
    if (i < HH * WC) atomicAdd(&cnt[radius[i]], 1.0f);
}

// ------------------------------------------------------------------- luma ---
__global__ void k_luma(const float* __restrict__ pred, float* __restrict__ gray, int b0) {
    int idx = blockIdx.x * 256 + threadIdx.x;
    if (idx >= CHUNK * HH * WW) return;
    int b = idx / (HH * WW);
    int p = idx - b * (HH * WW);
    const float* base = pred + (size_t)(b0 + b) * 3u * HH * WW;
    float r = base[p];
    float g = base[(size_t)HH * WW + p];
    float bl = base[2u * (size_t)HH * WW + p];
    gray[idx] = 0.299f * r + 0.587f * g + 0.114f * bl;
}

// ------------------------------------------------------- WMMA fragment ops --
// A fragment (16x4 f32): lane l holds row (l&15), K = {2*(l>>4), 2*(l>>4)+1}
// B fragment (4x16 f32): lane l holds col (l&15), K = {2*(l>>4), 2*(l>>4)+1}
// C/D (16x16 f32):       VGPR j -> row j + 8*(l>>4), col (l&15)

__device__ __forceinline__ v8f wmma_f32(v2f a, v2f b, v8f c) {
    return __builtin_amdgcn_wmma_f32_16x16x4_f32(
        false, a, false, b, (short)0, c, false, false);
}

// ----------------------------------------------- GEMM: real A x complex B ---
// Cr = A@Br, Ci = A@Bi.  A: [z][512x512]  B: [512xNP]  C: [z][512xNP]
__global__ void __launch_bounds__(256)
k_gemm_rc(const float* __restrict__ A,
          const float* __restrict__ Br, const float* __restrict__ Bi,
          float* __restrict__ Cr, float* __restrict__ Ci) {
    __shared__ __align__(16) float sA [2][64][APAD];
    __shared__ __align__(16) float sBr[2][32][APAD];
    __shared__ __align__(16) float sBi[2][32][APAD];

    int t = threadIdx.x, lane = t & 31, wv = t >> 5;
    int wm = wv >> 1, wn = wv & 1;                  // 4x2 waves of 16x16 tiles
    int rowBase = blockIdx.x * 64, colBase = blockIdx.y * 32;
    const float* Ab  = A  + (size_t)blockIdx.z * 512u * 512u;
    float*       Crb = Cr + (size_t)blockIdx.z * 512u * NP;
    float*       Cib = Ci + (size_t)blockIdx.z * 512u * NP;

    v8f accR = {}; v8f accI = {};
    int hi = lane >> 4, lo = lane & 15;

    // tile fill: A 64x32 = 512 xfloat4 (2/thread), each B 32x32 = 256 (1/thread)
    auto load_tiles = [&](int k0, int buf) {
#pragma unroll
        for (int i = 0; i < 2; i++) {
            int idx = t + i * 256;
            int r = idx >> 3, c4 = (idx & 7) * 4;
            cp16(Ab + (size_t)(rowBase + r) * 512u + k0 + c4, &sA[buf][r][c4]);
        }
        {
            int r = t >> 3, c4 = (t & 7) * 4;
            cp16(Br + (size_t)(k0 + r) * NP + colBase + c4, &sBr[buf][r][c4]);
            cp16(Bi + (size_t)(k0 + r) * NP + colBase + c4, &sBi[buf][r][c4]);
        }
    };

    load_tiles(0, 0);
    wait_async();
    __syncthreads();

    for (int k0 = 0; k0 < 512; k0 += 32) {
        int buf = (k0 >> 5) & 1;
        if (k0 + 32 < 512) load_tiles(k0 + 32, buf ^ 1);   // overlap next fetch
#pragma unroll
        for (int kk = 0; kk < 32; kk += 4) {
            v2f a, br, bi;
            a.x  = sA [buf][wm * 16 + lo][kk + 2 * hi];
            a.y  = sA [buf][wm * 16 + lo][kk + 2 * hi + 1];
            br.x = sBr[buf][kk + 2 * hi    ][wn * 16 + lo];
            br.y = sBr[buf][kk + 2 * hi + 1][wn * 16 + lo];
            bi.x = sBi[buf][kk + 2 * hi    ][wn * 16 + lo];
            bi.y = sBi[buf][kk + 2 * hi + 1][wn * 16 + lo];
            accR = wmma_f32(a, br, accR);
            accI = wmma_f32(a, bi, accI);
        }
        wait_async();
        __syncthreads();
    }
    int col = colBase + wn * 16 + lo;
#pragma unroll
    for (int j = 0; j < 8; j++) {
        int row = rowBase + wm * 16 + j + 8 * hi;
        Crb[(size_t)row * NP + col] = accR[j];
        Cib[(size_t)row * NP + col] = accI[j];
    }
}

// -------------------------------------------- GEMM: complex A x complex B ---
// Cr = Ar@Br - Ai@Bi ; Ci = Ar@Bi + Ai@Br.
// A (D matrix) shared across batches; B/C batched via blockIdx.z.
__global__ void __launch_bounds__(256)
k_gemm_cc(const float* __restrict__ Ar, const float* __restrict__ Ai,
          const float* __restrict__ Br, const float* __restrict__ Bi,
          float* __restrict__ Cr, float* __restrict__ Ci) {
    __shared__ __align__(16) float sAr[2][64][APAD];
    __shared__ __align__(16) float sAi[2][64][APAD];
    __shared__ __align__(16) float sBr[2][32][APAD];
    __shared__ __align__(16) float sBi[2][32][APAD];

    int t = threadIdx.x, lane = t & 31, wv = t >> 5;
    int wm = wv >> 1, wn = wv & 1;
    int rowBase = blockIdx.x * 64, colBase = blockIdx.y * 32;
    const float* Brb = Br + (size_t)blockIdx.z * 512u * NP;
    const float* Bib = Bi + (size_t)blockIdx.z * 512u * NP;
    float*       Crb = Cr + (size_t)blockIdx.z * 512u * NP;
    float*       Cib = Ci + (size_t)blockIdx.z * 512u * NP;

    v8f aP1 = {}; v8f aS1 = {}; v8f aP2 = {}; v8f aS2 = {};
    int hi = lane >> 4, lo = lane & 15;

    auto load_tiles = [&](int k0, int buf) {
#pragma unroll
        for (int i = 0; i < 2; i++) {
            int idx = t + i * 256;
            int r = idx >> 3, c4 = (idx & 7) * 4;
            cp16(Ar + (size_t)(rowBase + r) * 512u + k0 + c4, &sAr[buf][r][c4]);
            cp16(Ai + (size_t)(rowBase + r) * 512u + k0 + c4, &sAi[buf][r][c4]);
        }
        {
            int r = t >> 3, c4 = (t & 7) * 4;
            cp16(Brb + (size_t)(k0 + r) * NP + colBase + c4, &sBr[buf][r][c4]);
            cp16(Bib + (size_t)(k0 + r) * NP + colBase + c4, &sBi[buf][r][c4]);
        }
    };

    load_tiles(0, 0);
    wait_async();
    __syncthreads();

    for (int k0 = 0; k0 < 512; k0 += 32) {
        int buf = (k0 >> 5) & 1;
        if (k0 + 32 < 512) load_tiles(k0 + 32, buf ^ 1);
#pragma unroll
        for (int kk = 0; kk < 32; kk += 4) {
            v2f ar, ai, br, bi;
            ar.x = sAr[buf][wm * 16 + lo][kk + 2 * hi];
            ar.y = sAr[buf][wm * 16 + lo][kk + 2 * hi + 1];
            ai.x = sAi[buf][wm * 16 + lo][kk + 2 * hi];
            ai.y = sAi[buf][wm * 16 + lo][kk + 2 * hi + 1];
            br.x = sBr[buf][kk + 2 * hi    ][wn * 16 + lo];
            br.y = sBr[buf][kk + 2 * hi + 1][wn * 16 + lo];
            bi.x = sBi[buf][kk + 2 * hi    ][wn * 16 + lo];
            bi.y = sBi[buf][kk + 2 * hi + 1][wn * 16 + lo];
            aP1 = wmma_f32(ar, br, aP1);   // Ar@Br
            aS1 = wmma_f32(ai, bi, aS1);   // Ai@Bi
            aP2 = wmma_f32(ar, bi, aP2);   // Ar@Bi
            aS2 = wmma_f32(ai, br, aS2);   // Ai@Br
        }
        wait_async();
        __syncthreads();
    }
    int col = colBase + wn * 16 + lo;
#pragma unroll
    for (int j = 0; j < 8; j++) {
        int row = rowBase + wm * 16 + j + 8 * hi;
        Crb[(size_t)row * NP + col] = aP1[j] - aS1[j];
        Cib[(size_t)row * NP + col] = aP2[j] + aS2[j];
    }
}

// ----------------------------------------------- power spectrum + binning ---
__global__ void k_bin(const float* __restrict__ Fr, const float* __restrict__ Fi,
                      const int* __restrict__ radius, float* __restrict__ sums, int b0) {
    int idx = blockIdx.x * 256 + threadIdx.x;
    if (idx >= CHUNK * HH * WC) return;
    int b = idx / (HH * WC);
    int rem = idx - b * (HH * WC);
    int r = rem / WC;
    int k = rem - r * WC;
    size_t o = (size_t)b * 512u * NP + (size_t)r * NP + k;
    float re = Fr[o], im = Fi[o];
    float p = 20.0f * logf(re * re + im * im + 1e-8f);
    atomicAdd(&sums[(size_t)(b0 + b) * VLEN + radius[rem]], p);
}

// ------------------------------------------------------------- normalize ----
__global__ void k_finalize(const float* __restrict__ sums, const float* __restrict__ cnt,
                           float* __restrict__ out) {
    __shared__ float prof[VLEN];
    __shared__ float smn, smx;
    int b = blockIdx.x;
    for (int v = threadIdx.x; v < VLEN; v += blockDim.x)
        prof[v] = sums[(size_t)b * VLEN + v] / cnt[v];
    __syncthreads();
    if (threadIdx.x == 0) {
        float lo = prof[0], hi = prof[0];
        for (int v = 1; v < VLEN; v++) { lo = fminf(lo, prof[v]); hi = fmaxf(hi, prof[v]); }
        smn = lo; smx = hi;
    }
    __syncthreads();
    float inv = 1.0f / (smx - smn);
    for (int v = threadIdx.x; v < VLEN; v += blockDim.x)
        out[(size_t)b * VLEN + v] = (prof[v] - smn) * inv;
}

// ---------------------------------------------------------------- launch ----
extern "C" void kernel_launch(void* const* d_in, const int* in_sizes, int n_in,
                              void* d_out, int out_size, void* d_ws, size_t ws_size,
                              hipStream_t stream) {
    (void)in_sizes; (void)n_in; (void)out_size; (void)ws_size;
    const float* pred   = (const float*)d_in[0];
    const int*   radius = (const int*)d_in[1];
    float* ws = (float*)d_ws;

    float* Wr   = ws + OFF_WR;
    float* Wi   = ws + OFF_WI;
    float* Dr   = ws + OFF_DR;
    float* Di   = ws + OFF_DI;
    float* gray = ws + OFF_GRAY;
    float* F1r  = ws + OFF_F1R;
    float* F1i  = ws + OFF_F1I;
    float* Fr   = ws + OFF_FR;
    float* Fi   = ws + OFF_FI;
    float* sums = ws + OFF_SUMS;
    float* cnt  = ws + OFF_CNT;

    k_twiddle<<<dim3(32, 32), dim3(16, 16), 0, stream>>>(ws);
    {
        int n = NB * VLEN + VLEN;  // sums + counts contiguous
        k_init<<<(n + 255) / 256, 256, 0, stream>>>(sums, n);
    }
    k_counts<<<(HH * WC + 255) / 256, 256, 0, stream>>>(radius, cnt);

    for (int b0 = 0; b0 < NB; b0 += CHUNK) {
        k_luma<<<(CHUNK * HH * WW + 255) / 256, 256, 0, stream>>>(pred, gray, b0);
        // F1 = G @ W   (grid: M tiles x N tiles x batch)
        k_gemm_rc<<<dim3(512 / 64, NP / 32, CHUNK), 256, 0, stream>>>(gray, Wr, Wi, F1r, F1i);
        // F  = D @ F1
        k_gemm_cc<<<dim3(512 / 64, NP / 32, CHUNK), 256, 0, stream>>>(Dr, Di, F1r, F1i, Fr, Fi);
        k_bin<<<(CHUNK * HH * WC + 255) / 256, 256, 0, stream>>>(Fr, Fi, radius, sums, b0);
    }
    k_finalize<<<NB, 256, 0, stream>>>(sums, cnt, (float*)d_out);
}